// SoftDWT_84293028151705
// MI455X (gfx1250) — compile-verified
//
#include <hip/hip_runtime.h>
#include <math.h>

typedef __attribute__((ext_vector_type(2))) float v2f;
typedef __attribute__((ext_vector_type(8))) float v8f;

#define NSEQ   4096
#define NW     32            // waves per block (wave32)
#define NTHR   (NW * 32)     // 1024 threads
#define NTI    64            // row-tiles (64 rows each)
#define NTJ    256           // col-tiles (16 cols each)
#define NDIAG  (NTI + NTJ - 1)

#define BIGF   1.0e10f
// exp(z/gamma) = exp2(z * (1/gamma)*log2(e));  gamma*ln(S) = (gamma*ln2)*log2(S)
#define KEXP   (14.42695040888963f)   // 10 * log2(e)
#define CLOG   (0.06931471805599453f) // 0.1 * ln(2)

template <bool V> struct Bc { static constexpr bool value = V; };

// lane n <- lane n-1 within each 16-lane DPP row (row_shr:1); lanes 0/16 get 0
__device__ __forceinline__ float dpp_shr1(float v) {
    int r = __builtin_amdgcn_update_dpp(0, __float_as_int(v),
                                        0x111 /*row_shr:1*/, 0xf, 0xf, true);
    return __int_as_float(r);
}

// raw hardware transcendentals: exact here since softmin sum S >= 1
// (denormal-range addends are invisible against the exp2(0)=1 term)
__device__ __forceinline__ float hw_exp2(float x) { return __builtin_amdgcn_exp2f(x); }
__device__ __forceinline__ float hw_log2(float x) { return __builtin_amdgcn_logf(x); }

__global__ __launch_bounds__(NTHR) void sdtw_wavefront(const float* __restrict__ X,
                                                       const float* __restrict__ Y,
                                                       float* __restrict__ out) {
    __shared__ float Xs[NSEQ];
    __shared__ float Ys[NSEQ];
    __shared__ float Ht[NSEQ];      // Ht[j] = R[i0-1, j]   (bottom rows of tiles above)
    __shared__ float Vl[NSEQ];      // Vl[i] = R[i, j0-1]   (right columns)
    __shared__ float Ca[NTI];       // per tile-row corner R[i0-1, j0-1]
    __shared__ float Db[NW * 1024]; // per-wave D scratch: 2 tiles x 32 rows x 16 cols

    const int tid  = threadIdx.x;
    const int wid  = tid >> 5;
    const int lane = tid & 31;
    const int half = lane >> 4;     // 0: tile A, 1: tile B
    const int jl   = lane & 15;     // column within 16-wide tile
    const int g    = (wid << 1) | half;   // tile-row index ti (fixed per half-slot)
    const int i0   = g << 6;              // global row base of this half's tile-row
    const int wb   = wid << 10;           // wave's Db base
    const bool jl0  = (jl == 0);
    const bool jl15 = (jl == 15);

    // ---- init LDS ----
    for (int i = tid; i < NSEQ; i += NTHR) {
        Xs[i] = X[i];
        Ys[i] = Y[i];
        Ht[i] = BIGF;
        Vl[i] = BIGF;
    }
    if (tid < NTI) Ca[tid] = (tid == 0) ? 0.0f : BIGF;
    __syncthreads();

    // ---- wavefront over tile anti-diagonals ----
    for (int T = 0; T < NDIAG; ++T) {
        const int tjw = T - (wid << 1);              // tile-col of half 0 (wave-uniform)
        if ((unsigned)tjw <= 256u) {                 // wave has at least one valid half
            const int  tj    = T - g;                // per-half tile-col
            const bool valid = (unsigned)tj < 256u;
            const int  j0    = (valid ? tj : 0) << 4;

            // per-tile setup (reads happen before any write this diagonal)
            const float cornerVal   = Ca[g];
            const float topRightOld = Ht[j0 + 15];
            const float htj         = Ht[j0 + jl];
            const float htjm1       = Ht[j0 + (jl ? jl - 1 : 0)];

            // B matrices (constant per tile): rank-3 cost factorization
            // k0:(x^2)*(1)  k1:(x)*(-2y)  k2:(1)*(y^2)  k3:0
            const int jb0 = ((unsigned)tjw       < 256u ? tjw     : 0) << 4;
            const int jb1 = ((unsigned)(tjw - 1) < 256u ? tjw - 1 : 0) << 4;
            const float y0 = Ys[jb0 + jl];
            const float y1 = Ys[jb1 + jl];
            v2f B0, B1;
            B0.x = (half == 0) ? 1.0f : y0 * y0;    // lanes 0-15: K=0, 16-31: K=2
            B0.y = (half == 0) ? -2.0f * y0 : 0.0f; // lanes 0-15: K=1, 16-31: K=3
            B1.x = (half == 0) ? 1.0f : y1 * y1;
            B1.y = (half == 0) ? -2.0f * y1 : 0.0f;

            const int xb0 = (wid << 7);        // rows of tile A (g = 2*wid)
            const int xb1 = (wid << 7) + 64;   // rows of tile B

            float r1 = BIGF, r2 = BIGF, vlp = BIGF;

            // D chunk rows [16b,16b+16) for both tiles via one WMMA each (full EXEC)
            auto chunk = [&](int b) {
                const int rb = b << 4;
                {
                    const float xv = Xs[xb0 + rb + jl];
                    v2f A;
                    A.x = (half == 0) ? xv * xv : 1.0f;
                    A.y = (half == 0) ? xv : 0.0f;
                    v8f acc = __builtin_amdgcn_wmma_f32_16x16x4_f32(
                        false, A, false, B0, (short)0, (v8f){}, false, false);
#pragma unroll
                    for (int v = 0; v < 8; ++v) {
                        const int rl = rb + v + (half << 3);
                        Db[wb + ((rl & 31) << 4) + jl] = acc[v];
                    }
                }
                {
                    const float xv = Xs[xb1 + rb + jl];
                    v2f A;
                    A.x = (half == 0) ? xv * xv : 1.0f;
                    A.y = (half == 0) ? xv : 0.0f;
                    v8f acc = __builtin_amdgcn_wmma_f32_16x16x4_f32(
                        false, A, false, B1, (short)0, (v8f){}, false, false);
#pragma unroll
                    for (int v = 0; v < 8; ++v) {
                        const int rl = rb + v + (half << 3);
                        Db[wb + 512 + ((rl & 31) << 4) + jl] = acc[v];
                    }
                }
            };

            // 16 anti-diagonal micro-steps; FIRST: block 0 boundary selects,
            // HSTORE: bottom-row store possible (b>=3), VCHECK: r<64 guard (b==4)
            auto inner = [&](int bb, auto Fc, auto Hc, auto Vc) {
                constexpr bool FIRST  = decltype(Fc)::value;
                constexpr bool HSTORE = decltype(Hc)::value;
                constexpr bool VCHECK = decltype(Vc)::value;
#pragma unroll 4
                for (int s = 0; s < 16; ++s) {
                    const int sig = (bb << 4) + s;
                    const int r   = sig - jl;              // local row, cell (r, jl)
                    const float sh1 = dpp_shr1(r1);
                    const float sh2 = dpp_shr1(r2);
                    const float vl  = Vl[i0 + (sig < 64 ? sig : 63)];
                    const float d   = Db[wb + (half << 9) + (((unsigned)r & 31u) << 4) + jl];

                    float up, lf, dg;
                    if (FIRST) {
                        const bool rz = (r == 0);
                        up = rz ? htj : r1;
                        lf = jl0 ? vl : sh1;
                        dg = jl0 ? (sig == 0 ? cornerVal : vlp)
                                 : (rz ? htjm1 : sh2);
                    } else {
                        up = r1;
                        lf = jl0 ? vl : sh1;
                        dg = jl0 ? vlp : sh2;
                    }

                    const float m  = fminf(up, fminf(lf, dg));
                    const float S  = hw_exp2((m - up) * KEXP)
                                   + hw_exp2((m - lf) * KEXP)
                                   + hw_exp2((m - dg) * KEXP);
                    const float rn = d + m - CLOG * hw_log2(S);

                    if (valid) {
                        if (HSTORE) {
                            if (r == 63) Ht[j0 + jl] = rn;       // bottom row
                        }
                        if (FIRST) {
                            if (jl15 && r >= 0) Vl[i0 + r] = rn; // right col (r==0)
                        } else if (VCHECK) {
                            if (jl15 && r < 64) Vl[i0 + r] = rn;
                        } else {
                            if (jl15) Vl[i0 + r] = rn;           // always in-range
                        }
                    }
                    r2 = r1; r1 = rn; vlp = vl;
                }
            };

            chunk(0); inner(0, Bc<true>{},  Bc<false>{}, Bc<false>{});
            chunk(1); inner(1, Bc<false>{}, Bc<false>{}, Bc<false>{});
            chunk(2); inner(2, Bc<false>{}, Bc<false>{}, Bc<false>{});
            chunk(3); inner(3, Bc<false>{}, Bc<true>{},  Bc<false>{});
                      inner(4, Bc<false>{}, Bc<true>{},  Bc<true>{});

            if (valid && jl0) Ca[g] = topRightOld;  // corner for (g, tj+1)
        }
        __syncthreads();
    }

    if (tid == 0) out[0] = fabsf(Ht[NSEQ - 1]);   // |R[4095,4095]|
}

extern "C" void kernel_launch(void* const* d_in, const int* in_sizes, int n_in,
                              void* d_out, int out_size, void* d_ws, size_t ws_size,
                              hipStream_t stream) {
    (void)in_sizes; (void)n_in; (void)d_ws; (void)ws_size; (void)out_size;
    const float* x = (const float*)d_in[0];
    const float* y = (const float*)d_in[1];
    float* out = (float*)d_out;
    sdtw_wavefront<<<1, NTHR, 0, stream>>>(x, y, out);
}